// Mamba2_77446850282149
// MI455X (gfx1250) — compile-verified
//
#include <hip/hip_runtime.h>
#include <hip/hip_bf16.h>
#include <math.h>
#include <stdint.h>

// Fixed problem shape (from setup_inputs)
#define Bb      2
#define Lseq    4096
#define DMODEL  1024
#define DIN     2048            // d_inner
#define DSTATE  128
#define HDIM    64
#define NHEADS  32
#define KCONV   4
#define DPROJ   4384            // 2*DIN + 2*DSTATE + NHEADS (true width)
#define DPROJP  4416            // padded to 69*64 so GEMM1 has no N edge
#define CONVD   2304            // DIN + 2*DSTATE
#define CH      64              // chunk
#define NCHUNK  64              // Lseq / CH

#define KB      32              // GEMM K-block staged in LDS
#define LDA_S   36              // sA row stride (16B aligned, conflict-free)
#define LDB_S   68              // sB row stride (16B aligned)

typedef float v2f __attribute__((ext_vector_type(2)));
typedef float v8f __attribute__((ext_vector_type(8)));

// fp32 WMMA: D(16x16) = A(16x4) * B(4x16) + C   — CDNA5 V_WMMA_F32_16X16X4_F32
__device__ __forceinline__ v8f wmma4(v2f a, v2f b, v8f c) {
#if defined(__HIP_DEVICE_COMPILE__)
  return __builtin_amdgcn_wmma_f32_16x16x4_f32(false, a, false, b, (short)0, c,
                                               false, false);
#else
  return c;
#endif
}

// Async 16B global -> LDS copy (ASYNCcnt-tracked, §08 async tensor doc).
// Generic LDS pointer low 32 bits == wave-relative LDS byte address (ISA §10.2
// aperture truncation), which is exactly what VDST of the async op wants.
__device__ __forceinline__ void async_b128(const void* lds, const void* glob) {
#if defined(__HIP_DEVICE_COMPILE__)
  uint32_t lo = (uint32_t)(uintptr_t)lds;
  asm volatile("global_load_async_to_lds_b128 %0, %1, off"
               :: "v"(lo), "v"(glob) : "memory");
#else
  (void)lds; (void)glob;
#endif
}

__device__ __forceinline__ void wait_async0() {
#if defined(__HIP_DEVICE_COMPILE__)
  asm volatile("s_wait_asynccnt 0x0" ::: "memory");
#endif
}

__device__ __forceinline__ float sigmoidf_(float x) { return 1.f / (1.f + __expf(-x)); }

// ---------------------------------------------------------------------------
// Row-major GEMM C[M,N] = A[M,K] @ B[K,N], fp32 WMMA, async-LDS double buffer.
// Requires M%64==0, N%64==0, K%32==0. 64x64 block tile, 4 waves.
// ---------------------------------------------------------------------------
__global__ __launch_bounds__(128) void gemm_wmma_f32_async(
    const float* __restrict__ A, const float* __restrict__ B,
    float* __restrict__ C, int M, int N, int K, int lda, int ldb, int ldc)
{
  __shared__ float sA[2][64][LDA_S];
  __shared__ float sB[2][KB][LDB_S];

  const int tid  = threadIdx.x;
  const int w    = tid >> 5;
  const int lane = tid & 31;
  const int l16  = lane & 15;
  const int lh   = lane >> 4;
  const int m0   = blockIdx.x * 64;
  const int n0   = blockIdx.y * 64;

  auto issue = [&](int buf, int kk) {
#pragma unroll
    for (int i = 0; i < 4; ++i) {
      const int t  = tid + 128 * i;
      const int ar = t >> 3, ac = (t & 7) * 4;        // A: 64 rows x 32 cols
      async_b128(&sA[buf][ar][ac], A + (size_t)(m0 + ar) * lda + kk + ac);
      const int br = t >> 4, bc = (t & 15) * 4;       // B: 32 rows x 64 cols
      async_b128(&sB[buf][br][bc], B + (size_t)(kk + br) * ldb + n0 + bc);
    }
  };

  v8f acc[4] = {};
  const int rowA = w * 16 + l16;

  issue(0, 0);
  int buf = 0;
  for (int kk = 0; kk < K; kk += KB, buf ^= 1) {
    wait_async0();          // our async copies into `buf` are done
    __syncthreads();        // everyone's copies visible; prev reads of buf^1 done
    if (kk + KB < K) issue(buf ^ 1, kk + KB);
#pragma unroll
    for (int k4 = 0; k4 < KB; k4 += 4) {
      const int kb = k4 + 2 * lh;
      v2f a; a.x = sA[buf][rowA][kb]; a.y = sA[buf][rowA][kb + 1];
#pragma unroll
      for (int n = 0; n < 4; ++n) {
        const int cn = n * 16 + l16;
        v2f bv; bv.x = sB[buf][kb][cn]; bv.y = sB[buf][kb + 1][cn];
        acc[n] = wmma4(a, bv, acc[n]);
      }
    }
  }
#pragma unroll
  for (int n = 0; n < 4; ++n)
#pragma unroll
    for (int v = 0; v < 8; ++v)
      C[(size_t)(m0 + w * 16 + v + 8 * lh) * ldc + n0 + n * 16 + l16] = acc[n][v];
}

// Zero-padded copy of W_in: [1024,4384] -> [1024,4416]
__global__ void pad_win(const float* __restrict__ W, float* __restrict__ Wp)
{
  const int idx = blockIdx.x * blockDim.x + threadIdx.x;
  if (idx >= DMODEL * DPROJP) return;
  const int col = idx % DPROJP, row = idx / DPROJP;
  Wp[idx] = (col < DPROJ) ? W[(size_t)row * DPROJ + col] : 0.f;
}

// ---------------------------------------------------------------------------
// Depthwise causal conv (k=4) + SiLU on the xBC slice of (padded) zxbcdt.
// ---------------------------------------------------------------------------
__global__ void conv_silu_kernel(const float* __restrict__ zx,
                                 const float* __restrict__ cw,
                                 const float* __restrict__ cb,
                                 float* __restrict__ out)
{
  const int idx = blockIdx.x * blockDim.x + threadIdx.x;
  if (idx >= Bb * Lseq * CONVD) return;
  const int ch  = idx % CONVD;
  const int row = idx / CONVD;
  const int l   = row % Lseq;
  float acc = cb[ch];
#pragma unroll
  for (int k = 0; k < KCONV; ++k) {
    const int lp = l - (KCONV - 1) + k;
    if (lp >= 0)
      acc += cw[k * CONVD + ch] * zx[(size_t)(row - (l - lp)) * DPROJP + DIN + ch];
  }
  out[(size_t)row * CONVD + ch] = acc * sigmoidf_(acc);
}

// dt = softplus(raw + dt_bias);  dtA = -exp(A_log)*dt
__global__ void dt_kernel(const float* __restrict__ zx,
                          const float* __restrict__ dt_bias,
                          const float* __restrict__ A_log,
                          float* __restrict__ dtO, float* __restrict__ dtA)
{
  const int idx = blockIdx.x * blockDim.x + threadIdx.x;
  if (idx >= Bb * Lseq * NHEADS) return;
  const int h   = idx % NHEADS;
  const int row = idx / NHEADS;
  const float v  = zx[(size_t)row * DPROJP + 2 * DIN + 2 * DSTATE + h] + dt_bias[h];
  const float sp = (v > 20.f) ? v : log1pf(__expf(v));
  dtO[idx] = sp;
  dtA[idx] = -__expf(A_log[h]) * sp;
}

// ---------------------------------------------------------------------------
// Phase 1 per (b, chunk, head): Y_diag and per-chunk local states.
// ---------------------------------------------------------------------------
__global__ __launch_bounds__(128) void ssd_phase1(
    const float* __restrict__ xBC, const float* __restrict__ dt,
    const float* __restrict__ dtA, float* __restrict__ Y,
    float* __restrict__ states, float* __restrict__ Alast)
{
  __shared__ float sX[CH][HDIM + 4];   // x * dt
  __shared__ float sM[CH][CH + 4];     // masked decayed G
  __shared__ float sAc[CH];
  __shared__ float sDecay[CH];

  const int bid = blockIdx.x;
  const int h = bid % NHEADS;
  const int c = (bid / NHEADS) % NCHUNK;
  const int b = bid / (NHEADS * NCHUNK);
  const int tid = threadIdx.x;
  const size_t rowBase = (size_t)(b * Lseq + c * CH);
  const float* Bm = xBC + rowBase * CONVD + DIN;
  const float* Cm = Bm + DSTATE;

  for (int i = tid; i < CH * HDIM; i += 128) {
    const int l = i >> 6, p = i & 63;
    sX[l][p] = xBC[(rowBase + l) * CONVD + h * HDIM + p] *
               dt[(rowBase + l) * NHEADS + h];
  }
  if (tid == 0) {
    float a = 0.f;
    for (int l = 0; l < CH; ++l) { a += dtA[(rowBase + l) * NHEADS + h]; sAc[l] = a; }
    const float last = a;
    for (int l = 0; l < CH; ++l) sDecay[l] = __expf(last - sAc[l]);
    Alast[(b * NHEADS + h) * NCHUNK + c] = last;
  }
  __syncthreads();

  const int w = tid >> 5, lane = tid & 31, l16 = lane & 15, lh = lane >> 4;

  // G = Cm @ Bm^T  (64x64, K=128), causal decay mask -> sM
  {
    v8f g[4] = {};
    const int rA = w * 16 + l16;
    for (int kk = 0; kk < DSTATE; kk += 4) {
      const int kb = kk + 2 * lh;
      v2f a; a.x = Cm[(size_t)rA * CONVD + kb]; a.y = Cm[(size_t)rA * CONVD + kb + 1];
#pragma unroll
      for (int n = 0; n < 4; ++n) {
        const int s = n * 16 + l16;
        v2f bv; bv.x = Bm[(size_t)s * CONVD + kb]; bv.y = Bm[(size_t)s * CONVD + kb + 1];
        g[n] = wmma4(a, bv, g[n]);
      }
    }
#pragma unroll
    for (int n = 0; n < 4; ++n)
#pragma unroll
      for (int v = 0; v < 8; ++v) {
        const int r = w * 16 + v + 8 * lh;
        const int s = n * 16 + l16;
        sM[r][s] = (s <= r) ? g[n][v] * __expf(sAc[r] - sAc[s]) : 0.f;
      }
  }
  __syncthreads();

  // Y_diag = M @ X  (64x64, K=64)
  {
    v8f y[4] = {};
    const int rA = w * 16 + l16;
    for (int kk = 0; kk < CH; kk += 4) {
      const int kb = kk + 2 * lh;
      v2f a; a.x = sM[rA][kb]; a.y = sM[rA][kb + 1];
#pragma unroll
      for (int n = 0; n < 4; ++n) {
        v2f bv; bv.x = sX[kb][n * 16 + l16]; bv.y = sX[kb + 1][n * 16 + l16];
        y[n] = wmma4(a, bv, y[n]);
      }
    }
#pragma unroll
    for (int n = 0; n < 4; ++n)
#pragma unroll
      for (int v = 0; v < 8; ++v) {
        const int r = w * 16 + v + 8 * lh;
        Y[(rowBase + r) * DIN + h * HDIM + n * 16 + l16] = y[n][v];
      }
  }

  // S = (X*decay)^T @ Bm  (64x128, K=64)
  {
    v8f s8[8] = {};
    const int rP = w * 16 + l16;
    for (int kk = 0; kk < CH; kk += 4) {
      const int kb = kk + 2 * lh;
      v2f a; a.x = sX[kb][rP] * sDecay[kb]; a.y = sX[kb + 1][rP] * sDecay[kb + 1];
#pragma unroll
      for (int n = 0; n < 8; ++n) {
        const int cn = n * 16 + l16;
        v2f bv; bv.x = Bm[(size_t)kb * CONVD + cn]; bv.y = Bm[(size_t)(kb + 1) * CONVD + cn];
        s8[n] = wmma4(a, bv, s8[n]);
      }
    }
    const size_t sbase = ((size_t)((b * NCHUNK + c) * NHEADS + h)) * HDIM * DSTATE;
#pragma unroll
    for (int n = 0; n < 8; ++n)
#pragma unroll
      for (int v = 0; v < 8; ++v) {
        const int p = w * 16 + v + 8 * lh;
        states[sbase + (size_t)p * DSTATE + n * 16 + l16] = s8[n][v];
      }
  }
}

// ---------------------------------------------------------------------------
// Inter-chunk recurrence (in-place): states[c] becomes state ENTERING chunk c.
// ---------------------------------------------------------------------------
__global__ void ssd_scan(float* __restrict__ states, const float* __restrict__ Alast)
{
  const int idx = blockIdx.x * blockDim.x + threadIdx.x;
  const int per = NHEADS * HDIM * DSTATE;           // per (b, chunk)
  if (idx >= Bb * per) return;
  const int b = idx / per;
  const int r = idx % per;
  const int h = r / (HDIM * DSTATE);
  const size_t base = (size_t)b * NCHUNK * per + r;
  const float* Al = Alast + (b * NHEADS + h) * NCHUNK;
  float carry = 0.f;
  for (int c = 0; c < NCHUNK; ++c) {
    const size_t off = base + (size_t)c * per;
    const float S = states[off];
    states[off] = carry;                            // state entering chunk c
    carry = __expf(Al[c]) * carry + S;
  }
}

// ---------------------------------------------------------------------------
// Phase 2: Y += exp(Acum) * (C @ R^T) + xs * D
// ---------------------------------------------------------------------------
__global__ __launch_bounds__(128) void ssd_phase2(
    const float* __restrict__ xBC, const float* __restrict__ dtA,
    const float* __restrict__ states, const float* __restrict__ Dv,
    float* __restrict__ Y)
{
  __shared__ float sE[CH];
  const int bid = blockIdx.x;
  const int h = bid % NHEADS;
  const int c = (bid / NHEADS) % NCHUNK;
  const int b = bid / (NHEADS * NCHUNK);
  const int tid = threadIdx.x;
  const size_t rowBase = (size_t)(b * Lseq + c * CH);
  const float* Cm = xBC + rowBase * CONVD + DIN + DSTATE;
  const float* R  = states + ((size_t)((b * NCHUNK + c) * NHEADS + h)) * HDIM * DSTATE;

  if (tid == 0) {
    float a = 0.f;
    for (int l = 0; l < CH; ++l) { a += dtA[(rowBase + l) * NHEADS + h]; sE[l] = __expf(a); }
  }
  __syncthreads();

  const int w = tid >> 5, lane = tid & 31, l16 = lane & 15, lh = lane >> 4;
  v8f acc[4] = {};
  const int rA = w * 16 + l16;
  for (int kk = 0; kk < DSTATE; kk += 4) {
    const int kb = kk + 2 * lh;
    v2f a; a.x = Cm[(size_t)rA * CONVD + kb]; a.y = Cm[(size_t)rA * CONVD + kb + 1];
#pragma unroll
    for (int n = 0; n < 4; ++n) {
      const int cp = n * 16 + l16;
      v2f bv; bv.x = R[(size_t)cp * DSTATE + kb]; bv.y = R[(size_t)cp * DSTATE + kb + 1];
      acc[n] = wmma4(a, bv, acc[n]);
    }
  }
  const float Dh = Dv[h];
#pragma unroll
  for (int n = 0; n < 4; ++n)
#pragma unroll
    for (int v = 0; v < 8; ++v) {
      const int r = w * 16 + v + 8 * lh;
      const int p = n * 16 + l16;
      const size_t yi = (rowBase + r) * DIN + h * HDIM + p;
      const float xs = xBC[(rowBase + r) * CONVD + h * HDIM + p];
      Y[yi] += sE[r] * acc[n][v] + xs * Dh;
    }
}

// ---------------------------------------------------------------------------
// y = Y * silu(z); RMS-norm with norm_w (in-place on Y).
// ---------------------------------------------------------------------------
__global__ __launch_bounds__(256) void gate_norm(const float* __restrict__ zx,
                                                 const float* __restrict__ norm_w,
                                                 float* __restrict__ Y)
{
  __shared__ float red[256];
  const int row = blockIdx.x;
  const int tid = threadIdx.x;
  const float* zr = zx + (size_t)row * DPROJP;
  float* yr = Y + (size_t)row * DIN;
  float loc[8];
  float ss = 0.f;
#pragma unroll
  for (int j = 0; j < 8; ++j) {
    const int i = tid + j * 256;
    const float z = zr[i];
    const float v = yr[i] * (z * sigmoidf_(z));
    loc[j] = v;
    ss += v * v;
  }
  red[tid] = ss;
  __syncthreads();
  for (int s = 128; s > 0; s >>= 1) {
    if (tid < s) red[tid] += red[tid + s];
    __syncthreads();
  }
  const float rms = rsqrtf(red[0] / (float)DIN + 1e-5f);
#pragma unroll
  for (int j = 0; j < 8; ++j) {
    const int i = tid + j * 256;
    yr[i] = loc[j] * rms * norm_w[i];
  }
}

// ---------------------------------------------------------------------------
extern "C" void kernel_launch(void* const* d_in, const int* in_sizes, int n_in,
                              void* d_out, int out_size, void* d_ws, size_t ws_size,
                              hipStream_t stream)
{
  const float* x       = (const float*)d_in[0];
  const float* W_in    = (const float*)d_in[1];
  const float* conv_w  = (const float*)d_in[2];
  const float* conv_b  = (const float*)d_in[3];
  const float* dt_bias = (const float*)d_in[4];
  const float* A_log   = (const float*)d_in[5];
  const float* Dv      = (const float*)d_in[6];
  const float* norm_w  = (const float*)d_in[7];
  const float* W_out   = (const float*)d_in[8];
  float* out = (float*)d_out;

  float* ws = (float*)d_ws;
  size_t o = 0;
  float* Wp     = ws + o; o += (size_t)DMODEL * DPROJP;                      //  18 MB
  float* zx     = ws + o; o += (size_t)Bb * Lseq * DPROJP;                   // 145 MB
  float* xBCc   = ws + o; o += (size_t)Bb * Lseq * CONVD;                    //  75 MB
  float* dtB    = ws + o; o += (size_t)Bb * Lseq * NHEADS;                   //   1 MB
  float* dtA    = ws + o; o += (size_t)Bb * Lseq * NHEADS;                   //   1 MB
  float* Ybuf   = ws + o; o += (size_t)Bb * Lseq * DIN;                      //  67 MB
  float* states = ws + o; o += (size_t)Bb * NCHUNK * NHEADS * HDIM * DSTATE; // 134 MB
  float* Alast  = ws + o; o += (size_t)Bb * NHEADS * NCHUNK;

  // 0. pad W_in to N=4416 (removes all N-edge guards from GEMM1)
  pad_win<<<(DMODEL * DPROJP + 255) / 256, 256, 0, stream>>>(W_in, Wp);
  // 1. zxbcdt = x @ W_in   (padded N)
  gemm_wmma_f32_async<<<dim3((Bb * Lseq) / 64, DPROJP / 64), 128, 0, stream>>>(
      x, Wp, zx, Bb * Lseq, DPROJP, DMODEL, DMODEL, DPROJP, DPROJP);
  // 2. depthwise conv + SiLU
  conv_silu_kernel<<<(Bb * Lseq * CONVD + 255) / 256, 256, 0, stream>>>(
      zx, conv_w, conv_b, xBCc);
  // 3. dt / A*dt
  dt_kernel<<<(Bb * Lseq * NHEADS + 255) / 256, 256, 0, stream>>>(
      zx, dt_bias, A_log, dtB, dtA);
  // 4. intra-chunk SSD (Y_diag + local states)
  ssd_phase1<<<Bb * NCHUNK * NHEADS, 128, 0, stream>>>(
      xBCc, dtB, dtA, Ybuf, states, Alast);
  // 5. inter-chunk scan
  ssd_scan<<<(Bb * NHEADS * HDIM * DSTATE + 255) / 256, 256, 0, stream>>>(
      states, Alast);
  // 6. Y_off + skip connection
  ssd_phase2<<<Bb * NCHUNK * NHEADS, 128, 0, stream>>>(
      xBCc, dtA, states, Dv, Ybuf);
  // 7. gate + RMS norm
  gate_norm<<<Bb * Lseq, 256, 0, stream>>>(zx, norm_w, Ybuf);
  // 8. out = y @ W_out
  gemm_wmma_f32_async<<<dim3((Bb * Lseq) / 64, DMODEL / 64), 128, 0, stream>>>(
      Ybuf, W_out, out, Bb * Lseq, DMODEL, DIN, DIN, DMODEL, DMODEL);
}